// AssigmentLinear_70832600646121
// MI455X (gfx1250) — compile-verified
//
#include <hip/hip_runtime.h>

// out[b,a,o] = sum_i w[a,o,i]*x[b,a,i] + bias[a,o]
// B=4096, A=32, I=256, O=256 (fp32) -> V_WMMA_F32_16X16X4_F32
// Double-buffered LDS fed by GLOBAL_LOAD_ASYNC_TO_LDS_B128 (ASYNCcnt pipeline).

typedef float v2f __attribute__((ext_vector_type(2)));
typedef float v8f __attribute__((ext_vector_type(8)));
typedef int   v4i __attribute__((ext_vector_type(4)));

#define BSZ 4096
#define ASZ 32
#define ISZ 256
#define OSZ 256

#define MT  128        // M tile per workgroup
#define KC  32         // K chunk staged in LDS
#define LDP 36         // padded LDS row stride (floats): conflict-free b64 frag reads
#define NCHUNK (ISZ / KC)   // 8

#if defined(__AMDGCN__) && __has_builtin(__builtin_amdgcn_global_load_async_to_lds_b128) && __has_builtin(__builtin_amdgcn_s_wait_asynccnt)
#define USE_ASYNC 1
#else
#define USE_ASYNC 0
#endif

// Builtin prototype is (v4i addrspace(1)*, v4i addrspace(3)*, imm offset, imm cpol)
#define GPTR(p) ((__attribute__((address_space(1))) v4i*)(p))
#define LPTR(p) ((__attribute__((address_space(3))) v4i*)(p))

__global__ __launch_bounds__(256)
void AssigmentLinear_70832600646121_kernel(const float* __restrict__ x,
                                           const float* __restrict__ w,
                                           const float* __restrict__ bias,
                                           float* __restrict__ out)
{
#if USE_ASYNC
    __shared__ float As[2][MT * LDP];    // x tile:  [128 (b)][32 (i)]
    __shared__ float Bs[2][OSZ * LDP];   // w tile:  [256 (o)][32 (i)]
#else
    __shared__ float As[1][MT * LDP];
    __shared__ float Bs[1][OSZ * LDP];
#endif

    const int a      = blockIdx.y;
    const int m_base = blockIdx.x * MT;

    const int tid  = threadIdx.x;
    const int lane = tid & 31;
    const int wave = tid >> 5;
    const int wm   = wave >> 2;        // 0..1  -> M offset
    const int wn   = wave & 3;         // 0..3  -> N offset
    const int mW   = wm * 64;
    const int nW   = wn * 64;
    const int lr   = lane & 15;        // row/col within 16x16 tile
    const int kh   = (lane >> 4) * 2;  // K sub-offset: lanes 0-15 -> {0,1}, 16-31 -> {2,3}

    v8f acc[4][4] = {};                // 64x64 fp32 accumulators per wave

    const float* xa = x + (size_t)a * ISZ;              // b-stride = ASZ*ISZ
    const float* wa = w + (size_t)a * OSZ * ISZ;        // o-stride = ISZ

    const int r = tid >> 3;            // 0..31 : row handled by this thread
    const int c = (tid & 7) * 4;       // 0..28 : float4 column offset

    // ---- WMMA inner loop over one staged K chunk ----
    auto compute = [&](const float* __restrict__ Asb, const float* __restrict__ Bsb) {
        #pragma unroll
        for (int k = 0; k < KC; k += 4) {
            v2f af[4], bf[4];
            #pragma unroll
            for (int mi = 0; mi < 4; ++mi)
                af[mi] = *(const v2f*)(&Asb[(mW + mi * 16 + lr) * LDP + k + kh]);
            #pragma unroll
            for (int ni = 0; ni < 4; ++ni)
                bf[ni] = *(const v2f*)(&Bsb[(nW + ni * 16 + lr) * LDP + k + kh]);

            #pragma unroll
            for (int mi = 0; mi < 4; ++mi)
                #pragma unroll
                for (int ni = 0; ni < 4; ++ni)
                    acc[mi][ni] = __builtin_amdgcn_wmma_f32_16x16x4_f32(
                        false, af[mi], false, bf[ni],
                        (short)0, acc[mi][ni], false, false);
        }
    };

#if USE_ASYNC
    // ---- async DMA one K chunk into LDS buffer `buf` (12 b128 ops / wave) ----
    auto issue_chunk = [&](int k0, int buf) {
        #pragma unroll
        for (int it = 0; it < MT / 32; ++it) {
            const int row = r + it * 32;
            const float* g = xa + (size_t)(m_base + row) * (ASZ * ISZ) + k0 + c;
            __builtin_amdgcn_global_load_async_to_lds_b128(
                GPTR(g), LPTR(&As[buf][row * LDP + c]), 0, 0);
        }
        #pragma unroll
        for (int it = 0; it < OSZ / 32; ++it) {
            const int row = r + it * 32;
            const float* g = wa + (size_t)row * ISZ + k0 + c;
            __builtin_amdgcn_global_load_async_to_lds_b128(
                GPTR(g), LPTR(&Bs[buf][row * LDP + c]), 0, 0);
        }
    };

    issue_chunk(0, 0);
    for (int n = 0; n < NCHUNK - 1; ++n) {
        issue_chunk((n + 1) * KC, (n + 1) & 1);          // prefetch next chunk
        __builtin_amdgcn_s_wait_asynccnt(12);            // oldest 12 (chunk n) done
        __syncthreads();
        compute(&As[n & 1][0], &Bs[n & 1][0]);
        __syncthreads();                                  // done reading before overwrite
    }
    __builtin_amdgcn_s_wait_asynccnt(0);
    __syncthreads();
    compute(&As[(NCHUNK - 1) & 1][0], &Bs[(NCHUNK - 1) & 1][0]);
#else
    // ---- fallback: proven synchronous staging path ----
    for (int k0 = 0; k0 < ISZ; k0 += KC) {
        #pragma unroll
        for (int it = 0; it < MT / 32; ++it) {
            const int row = r + it * 32;
            float4 v = *(const float4*)(xa + (size_t)(m_base + row) * (ASZ * ISZ) + k0 + c);
            *(float4*)(&As[0][row * LDP + c]) = v;
        }
        #pragma unroll
        for (int it = 0; it < OSZ / 32; ++it) {
            const int row = r + it * 32;
            float4 v = *(const float4*)(wa + (size_t)row * ISZ + k0 + c);
            *(float4*)(&Bs[0][row * LDP + c]) = v;
        }
        __syncthreads();
        compute(&As[0][0], &Bs[0][0]);
        __syncthreads();
    }
#endif

    // ---- epilogue: D layout (VGPR rr: M = rr + 8*(lane/16), N = lane%16) ----
    const float* ba = bias + (size_t)a * OSZ;
    #pragma unroll
    for (int ni = 0; ni < 4; ++ni) {
        const int o   = nW + ni * 16 + lr;
        const float bn = ba[o];
        #pragma unroll
        for (int mi = 0; mi < 4; ++mi) {
            const int mrow = m_base + mW + mi * 16 + (lane >> 4) * 8;
            float* op = out + (size_t)mrow * (ASZ * OSZ) + (size_t)a * OSZ + o;
            #pragma unroll
            for (int rr = 0; rr < 8; ++rr)
                __builtin_nontemporal_store(acc[mi][ni][rr] + bn,
                                            op + (size_t)rr * (ASZ * OSZ));
        }
    }
}

extern "C" void kernel_launch(void* const* d_in, const int* in_sizes, int n_in,
                              void* d_out, int out_size, void* d_ws, size_t ws_size,
                              hipStream_t stream) {
    const float* x    = (const float*)d_in[0];   // [B, A, I]
    const float* wgt  = (const float*)d_in[1];   // [A, O, I]
    const float* bias = (const float*)d_in[2];   // [A, O]
    float* out        = (float*)d_out;           // [B, A, O]

    dim3 grid(BSZ / MT, ASZ);   // (32, 32)
    dim3 block(256);
    AssigmentLinear_70832600646121_kernel<<<grid, block, 0, stream>>>(x, wgt, bias, out);
}